// SelfAttention_21139829030941
// MI455X (gfx1250) — compile-verified
//
#include <hip/hip_runtime.h>
#include <hip/hip_bf16.h>
#include <math.h>

// Problem constants (fixed by the reference).
#define B_ 4
#define C_ 256
#define N_ 4096   // 64*64

typedef float v2f __attribute__((ext_vector_type(2)));
typedef float v8f __attribute__((ext_vector_type(8)));

// ---------------- WMMA f32 16x16x4 fragment helpers ----------------
// A is logical 16x4 (MxK), element (m,k) at p[m*rs + k*cs].
// ISA layout: lanes 0-15 -> K={0,1} in VGPR{0,1}; lanes 16-31 -> K={2,3}.
__device__ __forceinline__ v2f frag_a(const float* p, int rs, int cs, int lane) {
  int m = lane & 15;
  int k = (lane >> 4) << 1;
  v2f a;
  a.x = p[m * rs + k * cs];
  a.y = p[m * rs + (k + 1) * cs];
  return a;
}
// B is logical 4x16 (KxN), element (k,n) at p[k*ks + n*ns].
__device__ __forceinline__ v2f frag_b(const float* p, int ks, int ns, int lane) {
  int n = lane & 15;
  int k = (lane >> 4) << 1;
  v2f b;
  b.x = p[k * ks + n * ns];
  b.y = p[(k + 1) * ks + n * ns];
  return b;
}
// 8-arg form: (neg_a, A, neg_b, B, c_mod, C, reuse_a, reuse_b)
#define WMMA_F32(acc, a, b)                                                   \
  acc = __builtin_amdgcn_wmma_f32_16x16x4_f32(false, (a), false, (b),         \
                                              (short)0, (acc), false, false)

__device__ __forceinline__ float sigmoidf_(float x) {
  return 1.0f / (1.0f + __expf(-x));
}

// =====================================================================
// Kernel 1: fused projections.
//   q  = Wq x + bq   (gated)      -> qg  [B][N][C]  (row-major per pixel)
//   ge = geo Wgp^T + bgp          -> geb [B][N][C]
//   k  = Wk x + bk                -> kb  [B][C][N]
//   v  = Wv x + bv (transposed)   -> vT  [B][N][C]
// Block: 16 pixels x all 256 outputs; 8 waves x 2 o-tiles each.
// =====================================================================
__global__ void __launch_bounds__(256)
proj_kernel(const float* __restrict__ x, const float* __restrict__ geo,
            const float* __restrict__ Wq, const float* __restrict__ bq,
            const float* __restrict__ Wk, const float* __restrict__ bk,
            const float* __restrict__ Wv, const float* __restrict__ bv,
            const float* __restrict__ Wgp, const float* __restrict__ bgp,
            const float* __restrict__ Wgg, const float* __restrict__ bgg,
            float* __restrict__ qg, float* __restrict__ geb,
            float* __restrict__ kb, float* __restrict__ vT) {
  __shared__ float gate_acc[16];

  const int b    = blockIdx.y;
  const int n0   = blockIdx.x * 16;
  const int tid  = threadIdx.x;
  const int wave = tid >> 5;
  const int lane = tid & 31;
  const int nloc = lane & 15;
  const int mh   = (lane >> 4) ? 8 : 0;

  if (tid < 16) gate_acc[tid] = 0.0f;
  __syncthreads();

  const float* xb = x + (size_t)b * C_ * N_;   // [C][N]
  const float* gb = geo + (size_t)b * N_ * C_; // [N][C]

  const v8f zero = {0, 0, 0, 0, 0, 0, 0, 0};
  v8f aq[2], age[2], av[2], ak[2];
  for (int t = 0; t < 2; ++t) { aq[t] = zero; age[t] = zero; av[t] = zero; ak[t] = zero; }

  const int o0a = (wave * 2) * 16;

  for (int c0 = 0; c0 < C_; c0 += 4) {
    // A for q/v: x^T tile, (m,k) = x[(c0+k)*N + n0+m]
    v2f a_x  = frag_a(xb + n0 + (size_t)c0 * N_, 1, N_, lane);
    // A for ge: geo tile, (m,k) = geo[(n0+m)*C + c0+k]
    v2f a_ge = frag_a(gb + (size_t)n0 * C_ + c0, C_, 1, lane);
    // B for k: x tile, (k,n) = x[(c0+k)*N + n0+n]
    v2f b_x  = frag_b(xb + n0 + (size_t)c0 * N_, N_, 1, lane);
    for (int t = 0; t < 2; ++t) {
      const int o0 = o0a + t * 16;
      v2f b_wq  = frag_b(Wq  + (size_t)o0 * C_ + c0, 1, C_, lane); // (k,n)=Wq[(o0+n)*C + c0+k]
      v2f b_wgp = frag_b(Wgp + (size_t)o0 * C_ + c0, 1, C_, lane);
      v2f b_wv  = frag_b(Wv  + (size_t)o0 * C_ + c0, 1, C_, lane);
      v2f a_wk  = frag_a(Wk  + (size_t)o0 * C_ + c0, C_, 1, lane); // (m,k)=Wk[(o0+m)*C + c0+k]
      WMMA_F32(aq[t],  a_x,  b_wq);
      WMMA_F32(age[t], a_ge, b_wgp);
      WMMA_F32(av[t],  a_x,  b_wv);
      WMMA_F32(ak[t],  a_wk, b_x);
    }
  }

  // Biases.
  for (int t = 0; t < 2; ++t) {
    const int o0 = o0a + t * 16;
    const float vbq = bq[o0 + nloc], vbg = bgp[o0 + nloc], vbv = bv[o0 + nloc];
    for (int j = 0; j < 8; ++j) {
      aq[t][j]  += vbq;
      age[t][j] += vbg;
      av[t][j]  += vbv;
      ak[t][j]  += bk[o0 + j + mh];
    }
  }

  // Gate logits: sum_o Wgg[o]*q + Wgg[C+o]*ge per pixel (LDS float atomics).
  for (int t = 0; t < 2; ++t) {
    const int o = o0a + t * 16 + nloc;
    const float wgq = Wgg[o], wgg_ = Wgg[C_ + o];
    for (int j = 0; j < 8; ++j)
      atomicAdd(&gate_acc[j + mh], wgq * aq[t][j] + wgg_ * age[t][j]);
  }
  __syncthreads();

  const float bgg0 = bgg[0];
  for (int t = 0; t < 2; ++t) {
    const int o0 = o0a + t * 16;
    for (int j = 0; j < 8; ++j) {
      const int m = j + mh;
      const int n = n0 + m;
      const int o = o0 + nloc;
      const float gate = sigmoidf_(gate_acc[m] + bgg0);
      const float qv = aq[t][j] + gate * age[t][j];
      const size_t po = ((size_t)b * N_ + n) * C_ + o;
      qg[po]  = qv;
      geb[po] = age[t][j];
      vT[po]  = av[t][j];
      kb[((size_t)b * C_ + o0 + j + mh) * N_ + n0 + nloc] = ak[t][j];
    }
  }
}

// =====================================================================
// Kernel 2: fused energy -> softmax -> attention store -> att*vT -> gated out.
// Block owns 16 query rows; the full 16x4096 energy strip (256 KB) plus the
// 16x256 q strip (16 KB) live in the WGP's 320 KB LDS, so energies never
// round-trip HBM and the A operands of both GEMMs come from ds_load.
// =====================================================================
__global__ void __launch_bounds__(256)
attn_kernel(const float* __restrict__ qg, const float* __restrict__ kb,
            const float* __restrict__ vT, const float* __restrict__ geb,
            float* __restrict__ out, float* __restrict__ att) {
  extern __shared__ float smem[];
  float* e       = smem;                 // [16][N_]        256 KB
  float* qs      = smem + 16 * N_;       // [16][C_]         16 KB
  float* scratch = qs + 16 * C_;         // [256]
  float* rowmax  = scratch + 256;        // [16]
  float* rowsum  = rowmax + 16;          // [16]

  const int b    = blockIdx.y;
  const int m0   = blockIdx.x * 16;
  const int tid  = threadIdx.x;
  const int wave = tid >> 5;
  const int lane = tid & 31;
  const int nloc = lane & 15;
  const int mh   = (lane >> 4) ? 8 : 0;

  const float* qb  = qg + ((size_t)b * N_ + m0) * C_; // [16][C]
  const float* kbp = kb + (size_t)b * C_ * N_;        // [C][N]
  const float* vb  = vT + (size_t)b * N_ * C_;        // [N][C]

  const v8f zero = {0, 0, 0, 0, 0, 0, 0, 0};

  // Stage the 16x256 q strip into LDS (contiguous copy).
  for (int idx = tid; idx < 16 * C_; idx += 256) qs[idx] = qb[idx];
  __syncthreads();

  // ---- Phase 1: energy = q . k. Each wave does pairs of 16-wide column
  // tiles sharing one A fragment per k-step (A from LDS, B streams from L2).
  for (int i = 0; i < 32; i += 2) {
    const int nc0 = (wave + 8 * i) * 16;
    const int nc1 = (wave + 8 * (i + 1)) * 16;
    v8f acc0 = zero, acc1 = zero;
    for (int c0 = 0; c0 < C_; c0 += 4) {
      v2f a  = frag_a(qs + c0, C_, 1, lane);                       // (m,k)=q[m][c0+k]
      v2f b0 = frag_b(kbp + (size_t)c0 * N_ + nc0, N_, 1, lane);   // (k,n)=k[c0+k][nc+n]
      v2f b1 = frag_b(kbp + (size_t)c0 * N_ + nc1, N_, 1, lane);
      WMMA_F32(acc0, a, b0);
      WMMA_F32(acc1, a, b1);
    }
    for (int j = 0; j < 8; ++j) {
      e[(j + mh) * N_ + nc0 + nloc] = acc0[j];
      e[(j + mh) * N_ + nc1 + nloc] = acc1[j];
    }
  }
  __syncthreads();

  // ---- Phase 2: row softmax stats (16 threads per row) ----
  const int r   = tid >> 4;
  const int cst = tid & 15;
  float mx = -3.4e38f;
  for (int n = cst; n < N_; n += 16) mx = fmaxf(mx, e[r * N_ + n]);
  scratch[tid] = mx;
  __syncthreads();
  if (cst == 0) {
    float m2 = scratch[tid];
    for (int i = 1; i < 16; ++i) m2 = fmaxf(m2, scratch[tid + i]);
    rowmax[r] = m2;
  }
  __syncthreads();
  const float rmx = rowmax[r];
  float s = 0.0f;
  for (int n = cst; n < N_; n += 16) {
    const float ev = __expf(e[r * N_ + n] - rmx);
    e[r * N_ + n] = ev;   // keep unnormalized exp in LDS
    s += ev;
  }
  scratch[tid] = s;
  __syncthreads();
  if (cst == 0) {
    float s2 = 0.0f;
    for (int i = 0; i < 16; ++i) s2 += scratch[tid + i];
    rowsum[r] = s2;
  }
  __syncthreads();

  // ---- Phase 3: single normalized attention store (contiguous bursts) ----
  float* attb = att + (size_t)b * N_ * N_ + (size_t)m0 * N_;
  for (int idx = tid; idx < 16 * N_; idx += 256) {
    const int rr = idx >> 12;  // idx / 4096
    attb[idx] = e[idx] * (1.0f / rowsum[rr]);
  }

  // ---- Phase 4: out_T[m][c] = sum_n att[m][n] * vT[n][c].
  // A from LDS; both 16-wide o-tiles of this wave share each A fragment.
  {
    const int o0 = wave * 32;
    v8f acc0 = zero, acc1 = zero;
    for (int k0 = 0; k0 < N_; k0 += 4) {
      v2f a  = frag_a(e + k0, N_, 1, lane);                           // (m,k)=e[m][k0+k]
      v2f b0 = frag_b(vb + (size_t)k0 * C_ + o0, C_, 1, lane);        // (k,n)=vT[k0+k][o0+n]
      v2f b1 = frag_b(vb + (size_t)k0 * C_ + o0 + 16, C_, 1, lane);
      WMMA_F32(acc0, a, b0);
      WMMA_F32(acc1, a, b1);
    }
    for (int t = 0; t < 2; ++t) {
      const v8f acc = t ? acc1 : acc0;
      for (int j = 0; j < 8; ++j) {
        const int m = j + mh;
        const int c = o0 + t * 16 + nloc;
        const int p = m0 + m;
        const float val = acc[j] / rowsum[m];  // fold softmax normalization here
        // ge_r = ge.view(B,C,W,H): flat [b][c*N + p]
        const float g = geb[(size_t)b * N_ * C_ + (size_t)c * N_ + p];
        out[(size_t)b * C_ * N_ + (size_t)c * N_ + p] = val * (1.0f + sigmoidf_(g));
      }
    }
  }
}

// =====================================================================
extern "C" void kernel_launch(void* const* d_in, const int* in_sizes, int n_in,
                              void* d_out, int out_size, void* d_ws, size_t ws_size,
                              hipStream_t stream) {
  const float* x   = (const float*)d_in[0];
  const float* geo = (const float*)d_in[1];
  const float* Wq  = (const float*)d_in[2];
  const float* bq  = (const float*)d_in[3];
  const float* Wk  = (const float*)d_in[4];
  const float* bk  = (const float*)d_in[5];
  const float* Wv  = (const float*)d_in[6];
  const float* bv  = (const float*)d_in[7];
  const float* Wgp = (const float*)d_in[8];
  const float* bgp = (const float*)d_in[9];
  const float* Wgg = (const float*)d_in[10];
  const float* bgg = (const float*)d_in[11];

  float* outp = (float*)d_out;                  // out [B,C,W,H] first
  float* attp = outp + (size_t)B_ * C_ * N_;    // attention [B,N,N] second

  // Workspace: qg, ge, k, vT : 4 x B*N*C floats = 67 MB total (fits L2).
  const size_t BNC = (size_t)B_ * N_ * C_;
  float* qg  = (float*)d_ws;
  float* geb = qg + BNC;
  float* kbf = geb + BNC;
  float* vTf = kbf + BNC;

  dim3 grid(N_ / 16, B_);
  dim3 blk(256);

  proj_kernel<<<grid, blk, 0, stream>>>(x, geo, Wq, bq, Wk, bk, Wv, bv,
                                        Wgp, bgp, Wgg, bgg, qg, geb, kbf, vTf);

  const size_t smem = (size_t)(16 * N_ + 16 * C_ + 256 + 32) * sizeof(float); // ~273 KB
  hipFuncSetAttribute(reinterpret_cast<const void*>(attn_kernel),
                      hipFuncAttributeMaxDynamicSharedMemorySize, (int)smem);
  attn_kernel<<<grid, blk, smem, stream>>>(qg, kbf, vTf, geb, outp, attp);
}